// sp_Attn_head_41283225649259
// MI455X (gfx1250) — compile-verified
//
#include <hip/hip_runtime.h>
#include <hip/hip_bf16.h>
#include <math.h>

typedef float v2f __attribute__((ext_vector_type(2)));
typedef float v8f __attribute__((ext_vector_type(8)));

#define ALPHA 0.01f

// ---------------------------------------------------------------------------
// K0: zero u32 array
// ---------------------------------------------------------------------------
__global__ void zero_u32_kernel(unsigned* __restrict__ p, int n) {
    int i = blockIdx.x * blockDim.x + threadIdx.x;
    if (i < n) p[i] = 0u;
}

// ---------------------------------------------------------------------------
// K1a: transpose Wfc[128,256] -> WfcT[256,128] (k-major) so the GEMM's
// B-matrix loads are lane-coalesced instead of 1KB-strided.
// ---------------------------------------------------------------------------
__global__ __launch_bounds__(256) void transpose_wfc_kernel(
    const float* __restrict__ Wfc, float* __restrict__ WfcT)
{
    __shared__ float tile[32][33];
    const int kbase = blockIdx.x * 32;            // 0..224
    const int dbase = blockIdx.y * 32;            // 0..96
    const int tx = threadIdx.x, ty = threadIdx.y; // blockDim = (32, 8)
#pragma unroll
    for (int j = 0; j < 4; ++j)
        tile[ty + 8 * j][tx] = Wfc[(size_t)(dbase + ty + 8 * j) * 256 + kbase + tx];
    __syncthreads();
#pragma unroll
    for (int j = 0; j < 4; ++j)
        WfcT[(size_t)(kbase + ty + 8 * j) * 128 + dbase + tx] = tile[tx][ty + 8 * j];
}

// ---------------------------------------------------------------------------
// K1b: h[N,128] = xt[N,256] @ Wfc^T   via V_WMMA_F32_16X16X4_F32
// x is stored [256, N] (xt column-major -> coalesced A loads).
// Each wave: one 32-node supertile (2 M-tiles) x all 8 d-tiles
//   -> 16 x v8f accumulators, each B register pair feeds 2 WMMAs.
// A layout (16x4 f32): lane = 16*hi + m; VGPR0 = K=2*hi, VGPR1 = K=2*hi+1, M=m
// B layout (4x16 f32): lane = 16*hi + c; VGPR0 = K=2*hi, VGPR1 = K=2*hi+1, N=c
// C/D layout: VGPR r -> M = r + 8*hi, N = lane&15
// ---------------------------------------------------------------------------
__global__ __launch_bounds__(128) void gemm_h_kernel(
    const float* __restrict__ x,     // [256, N]
    const float* __restrict__ WfcT,  // [256, 128] k-major
    float* __restrict__ h,           // [N, 128]
    int N)
{
    const int wave = threadIdx.x >> 5;
    const int lane = threadIdx.x & 31;
    const int t32  = blockIdx.x * 4 + wave;      // 32-node supertile index
    const int nst  = N >> 5;
    if (t32 >= nst) return;                       // wave-uniform: EXEC all-1s

    const int m  = lane & 15;
    const int hi = lane >> 4;                     // 0 or 1

    v8f acc0[8] = {};
    v8f acc1[8] = {};

    const size_t n0 = (size_t)t32 * 32 + m;       // subtile-0 node for A loads
    const size_t n1 = n0 + 16;                    // subtile-1 node

    for (int k = 0; k < 256; k += 4) {
        const size_t xoff = (size_t)(k + 2 * hi) * N;
        v2f a0, a1;
        a0.x = x[xoff + n0];
        a0.y = x[xoff + N + n0];
        a1.x = x[xoff + n1];
        a1.y = x[xoff + N + n1];
        const size_t boff = (size_t)(k + 2 * hi) * 128 + m;
#pragma unroll
        for (int dt = 0; dt < 8; ++dt) {
            v2f b;
            b.x = WfcT[boff + dt * 16];           // coalesced: lanes stride 4B
            b.y = WfcT[boff + 128 + dt * 16];
            acc0[dt] = __builtin_amdgcn_wmma_f32_16x16x4_f32(
                false, a0, false, b, (short)0, acc0[dt], false, false);
            acc1[dt] = __builtin_amdgcn_wmma_f32_16x16x4_f32(
                false, a1, false, b, (short)0, acc1[dt], false, false);
        }
    }

#pragma unroll
    for (int dt = 0; dt < 8; ++dt) {
#pragma unroll
        for (int r = 0; r < 8; ++r) {
            const size_t node0 = (size_t)t32 * 32 + r + 8 * hi;
            h[node0 * 128 + dt * 16 + m]          = acc0[dt][r];
            h[(node0 + 16) * 128 + dt * 16 + m]   = acc1[dt][r];
        }
    }
}

// ---------------------------------------------------------------------------
// K2: per-node scalars a_st = h.w1 + b1 ; a_ed = h.w2 + b2   (one wave/node)
// ---------------------------------------------------------------------------
__global__ __launch_bounds__(256) void scalars_kernel(
    const float* __restrict__ h, const float* __restrict__ w1,
    const float* __restrict__ b1, const float* __restrict__ w2,
    const float* __restrict__ b2, float* __restrict__ a_st,
    float* __restrict__ a_ed, int N)
{
    const int wave = (blockIdx.x * blockDim.x + threadIdx.x) >> 5;
    const int lane = threadIdx.x & 31;
    if (wave >= N) return;

    const float* hr = h + (size_t)wave * 128;
    float s1 = 0.f, s2 = 0.f;
#pragma unroll
    for (int r = 0; r < 4; ++r) {
        const float hv = hr[lane + 32 * r];
        s1 += hv * w1[lane + 32 * r];
        s2 += hv * w2[lane + 32 * r];
    }
#pragma unroll
    for (int off = 16; off > 0; off >>= 1) {
        s1 += __shfl_xor(s1, off, 32);
        s2 += __shfl_xor(s2, off, 32);
    }
    if (lane == 0) {
        a_st[wave] = s1 + b1[0];
        a_ed[wave] = s2 + b2[0];
    }
}

// ---------------------------------------------------------------------------
// K3: degree count per destination row
// ---------------------------------------------------------------------------
__global__ void degree_kernel(const long long* __restrict__ ei,
                              unsigned* __restrict__ deg, int E)
{
    int e = blockIdx.x * blockDim.x + threadIdx.x;
    if (e >= E) return;
    atomicAdd(&deg[(int)ei[e]], 1u);
}

// ---------------------------------------------------------------------------
// K4: single-block exclusive scan -> rowptr[N+1], also init write cursors
// ---------------------------------------------------------------------------
__global__ __launch_bounds__(1024) void scan_kernel(
    const unsigned* __restrict__ deg, unsigned* __restrict__ rowptr,
    unsigned* __restrict__ wpos, int N)
{
    __shared__ unsigned buf[1024];
    __shared__ unsigned carry;
    const int tid = threadIdx.x;
    if (tid == 0) carry = 0;
    __syncthreads();

    for (int base = 0; base < N; base += 1024) {
        const int i = base + tid;
        const unsigned v = (i < N) ? deg[i] : 0u;
        buf[tid] = v;
        __syncthreads();
        for (int off = 1; off < 1024; off <<= 1) {
            const unsigned t = (tid >= off) ? buf[tid - off] : 0u;
            __syncthreads();
            buf[tid] += t;
            __syncthreads();
        }
        const unsigned excl = buf[tid] - v;
        const unsigned c = carry;                 // read before update
        if (i < N) { rowptr[i] = c + excl; wpos[i] = c + excl; }
        __syncthreads();
        if (tid == 1023) carry = c + buf[1023];
        __syncthreads();
    }
    if (tid == 0) rowptr[N] = carry;
}

// ---------------------------------------------------------------------------
// K5: scatter edges into CSR order, fusing edge score e = leakyrelu(ast+aed)
// ---------------------------------------------------------------------------
__global__ void scatter_kernel(const long long* __restrict__ ei,
                               const float* __restrict__ a_st,
                               const float* __restrict__ a_ed,
                               unsigned* __restrict__ wpos,
                               int* __restrict__ csr_col,
                               float* __restrict__ csr_e, int E)
{
    int e = blockIdx.x * blockDim.x + threadIdx.x;
    if (e >= E) return;
    const int row = (int)ei[e];
    const int col = (int)ei[E + e];
    const unsigned p = atomicAdd(&wpos[row], 1u);
    float ev = a_st[row] + a_ed[col];
    ev = ev > 0.f ? ev : ALPHA * ev;
    csr_col[p] = col;
    csr_e[p]   = ev;
}

// ---------------------------------------------------------------------------
// K6: per-node softmax + weighted aggregation + bias + ELU (one wave/node)
// Phase 1/2: lanes parallel over edges (max, exp-sum). Phase 3: edges
// sequential, lanes parallel over D=128 as float4 (512B coalesced gathers of
// h[col] — h is 51MB so this mostly hits the 192MB L2).
// ---------------------------------------------------------------------------
__global__ __launch_bounds__(256) void aggregate_kernel(
    const float* __restrict__ h, const unsigned* __restrict__ rowptr,
    const int* __restrict__ csr_col, const float* __restrict__ csr_e,
    const float* __restrict__ bias, float* __restrict__ ret, int N)
{
    const int wave = (blockIdx.x * blockDim.x + threadIdx.x) >> 5;
    const int lane = threadIdx.x & 31;
    if (wave >= N) return;

    const unsigned lo  = rowptr[wave];
    const unsigned hiE = rowptr[wave + 1];

    // phase 1: segment max
    float mx = -INFINITY;
    for (unsigned j = lo + lane; j < hiE; j += 32) mx = fmaxf(mx, csr_e[j]);
#pragma unroll
    for (int off = 16; off > 0; off >>= 1) mx = fmaxf(mx, __shfl_xor(mx, off, 32));

    // phase 2: exp-sum
    float s = 0.f;
    for (unsigned j = lo + lane; j < hiE; j += 32) s += __expf(csr_e[j] - mx);
#pragma unroll
    for (int off = 16; off > 0; off >>= 1) s += __shfl_xor(s, off, 32);
    const float inv = (hiE > lo) ? 1.f / s : 0.f;

    // phase 3: ret[row] = sum coef * h[col]
    float4 acc = make_float4(0.f, 0.f, 0.f, 0.f);
    for (unsigned j = lo; j < hiE; ++j) {
        const float coef = __expf(csr_e[j] - mx) * inv;
        const int c = csr_col[j];
        const float4 hv = ((const float4*)(h + (size_t)c * 128))[lane];
        acc.x += coef * hv.x; acc.y += coef * hv.y;
        acc.z += coef * hv.z; acc.w += coef * hv.w;
    }

    const float4 bv = ((const float4*)bias)[lane];
    acc.x += bv.x; acc.y += bv.y; acc.z += bv.z; acc.w += bv.w;
    acc.x = acc.x > 0.f ? acc.x : expm1f(acc.x);
    acc.y = acc.y > 0.f ? acc.y : expm1f(acc.y);
    acc.z = acc.z > 0.f ? acc.z : expm1f(acc.z);
    acc.w = acc.w > 0.f ? acc.w : expm1f(acc.w);

    ((float4*)(ret + (size_t)wave * 128))[lane] = acc;
}

// ---------------------------------------------------------------------------
// K7: transpose ret[N,128] -> out[128,N] with 32x32 LDS tiles (N % 32 == 0)
// ---------------------------------------------------------------------------
__global__ __launch_bounds__(256) void transpose_kernel(
    const float* __restrict__ ret, float* __restrict__ out, int N)
{
    __shared__ float tile[32][33];
    const int nbase = blockIdx.x * 32;
    const int dbase = blockIdx.y * 32;
    const int tx = threadIdx.x, ty = threadIdx.y;   // blockDim = (32, 8)
#pragma unroll
    for (int k = 0; k < 4; ++k)
        tile[ty + 8 * k][tx] =
            ret[(size_t)(nbase + ty + 8 * k) * 128 + dbase + tx];
    __syncthreads();
#pragma unroll
    for (int k = 0; k < 4; ++k)
        out[(size_t)(dbase + ty + 8 * k) * N + nbase + tx] = tile[tx][ty + 8 * k];
}

// ---------------------------------------------------------------------------
// launch
// ---------------------------------------------------------------------------
extern "C" void kernel_launch(void* const* d_in, const int* in_sizes, int n_in,
                              void* d_out, int out_size, void* d_ws, size_t ws_size,
                              hipStream_t stream)
{
    const float*     x    = (const float*)d_in[0];      // [1,256,N]
    const long long* ei   = (const long long*)d_in[1];  // [2,E] int64
    const float*     Wfc  = (const float*)d_in[2];      // [128,256]
    const float*     w1   = (const float*)d_in[3];
    const float*     b1   = (const float*)d_in[4];
    const float*     w2   = (const float*)d_in[5];
    const float*     b2   = (const float*)d_in[6];
    const float*     bias = (const float*)d_in[7];
    float*           out  = (float*)d_out;              // [1,128,N]

    const int N = in_sizes[0] / 256;
    const int E = in_sizes[1] / 2;

    // workspace carve-up (all 256B aligned)
    char* w = (char*)d_ws;
    auto align = [](size_t v) { return (v + 255) & ~(size_t)255; };
    float*    h       = (float*)w;     w += align((size_t)N * 128 * 4);
    float*    ret     = (float*)w;     w += align((size_t)N * 128 * 4);
    float*    WfcT    = (float*)w;     w += align((size_t)256 * 128 * 4);
    float*    a_st    = (float*)w;     w += align((size_t)N * 4);
    float*    a_ed    = (float*)w;     w += align((size_t)N * 4);
    unsigned* deg     = (unsigned*)w;  w += align((size_t)N * 4);
    unsigned* rowptr  = (unsigned*)w;  w += align((size_t)(N + 1) * 4);
    unsigned* wpos    = (unsigned*)w;  w += align((size_t)N * 4);
    int*      csr_col = (int*)w;       w += align((size_t)E * 4);
    float*    csr_e   = (float*)w;     w += align((size_t)E * 4);

    zero_u32_kernel<<<(N + 255) / 256, 256, 0, stream>>>(deg, N);

    transpose_wfc_kernel<<<dim3(8, 4), dim3(32, 8), 0, stream>>>(Wfc, WfcT);

    const int nst = N / 32;
    gemm_h_kernel<<<(nst + 3) / 4, 128, 0, stream>>>(x, WfcT, h, N);

    scalars_kernel<<<(N * 32 + 255) / 256, 256, 0, stream>>>(
        h, w1, b1, w2, b2, a_st, a_ed, N);

    degree_kernel<<<(E + 255) / 256, 256, 0, stream>>>(ei, deg, E);

    scan_kernel<<<1, 1024, 0, stream>>>(deg, rowptr, wpos, N);

    scatter_kernel<<<(E + 255) / 256, 256, 0, stream>>>(
        ei, a_st, a_ed, wpos, csr_col, csr_e, E);

    aggregate_kernel<<<(N * 32 + 255) / 256, 256, 0, stream>>>(
        h, rowptr, csr_col, csr_e, bias, ret, N);

    transpose_kernel<<<dim3(N / 32, 4), dim3(32, 8), 0, stream>>>(ret, out, N);
}